// PaiNN_24618752541188
// MI455X (gfx1250) — compile-verified
//
#include <hip/hip_runtime.h>
#include <math.h>

typedef float v2f __attribute__((ext_vector_type(2)));
typedef float v8f __attribute__((ext_vector_type(8)));

#define F_DIM   128
#define F3      384
#define NPM     32
#define NMOL    128
#define NATOMS  4096
#define N_RBF   20
#define CUTOFF  5.0f
#define CUTOFF2 25.0f
#define PI_F    3.14159265358979323846f

__device__ __forceinline__ float silu_f(float x) { return x / (1.0f + __expf(-x)); }

// ---------------------------------------------------------------------------
// WMMA f32 GEMM:  C[M x N] = act(A[M x K] @ W[K x N] + bias)
//
// Block = 128 threads (4 waves) -> 64x64 output tile.
//   * B tile (K x 64) staged once in LDS, shared by all 4 waves.
//     LDS layout is K-pair interleaved: element (k,n) lives at
//       shB[(k/2)*128 + 2*n + (k&1)]
//     so the WMMA B fragment (B[k][n], B[k+1][n]) is one contiguous,
//     8-byte-aligned float2 -> a single conflict-free ds_load_b64.
//   * Each wave owns 16 rows and all 64 cols: 4 accumulators, so one A
//     float2 fragment feeds 4 independent v_wmma_f32_16x16x4_f32 chains
//     (pipelines past the D->C accumulation latency).
// A-fragment layout (ISA 16x4): lanes 0-15 hold K={k0,k0+1}, lanes 16-31
// hold K={k0+2,k0+3} for row M=lane&15 -> contiguous float2 per lane.
// C/D layout: VGPR r -> row r (lanes 0-15) / row r+8 (lanes 16-31), col=lane&15.
// ---------------------------------------------------------------------------
__global__ __launch_bounds__(128)
void gemm_wmma_f32(const float* __restrict__ A, int lda,
                   const float* __restrict__ W, int ldb,
                   const float* __restrict__ bias,
                   float* __restrict__ C, int ldc,
                   int K, int act)
{
  extern __shared__ float shB[];           // K/2 pair-rows x 128

  const int tid  = threadIdx.x;
  const int wave = tid >> 5;
  const int lane = tid & 31;
  const int m    = lane & 15;
  const int half = lane >> 4;              // K sub-pair select
  const int nb   = blockIdx.x * 64;        // col base of block tile
  const int mb   = blockIdx.y * 64 + wave * 16;  // row base of this wave

  // cooperative stage of B tile into LDS, pair-interleaved (K*64 % 128 == 0)
  for (int idx = tid; idx < K * 64; idx += 128) {
    int k = idx >> 6, n = idx & 63;
    shB[((k >> 1) << 7) + 2 * n + (k & 1)] = W[(size_t)k * ldb + nb + n];
  }
  __syncthreads();

  const float* __restrict__ arow = A + (size_t)(mb + m) * lda + 2 * half;

  v8f acc0 = {}, acc1 = {}, acc2 = {}, acc3 = {};
  for (int k0 = 0; k0 < K; k0 += 4) {
    v2f a = *(const v2f*)(arow + k0);                 // A[m][k0+2h .. k0+2h+1]
    // pair-row for K rows {k0+2h, k0+2h+1}
    const float* bb = shB + (size_t)((k0 >> 1) + half) * 128;
    v2f b0 = *(const v2f*)(bb + 2 * m);               // cols nb+m
    v2f b1 = *(const v2f*)(bb + 2 * (16 + m));        // cols nb+16+m
    v2f b2 = *(const v2f*)(bb + 2 * (32 + m));        // cols nb+32+m
    v2f b3 = *(const v2f*)(bb + 2 * (48 + m));        // cols nb+48+m
    acc0 = __builtin_amdgcn_wmma_f32_16x16x4_f32(false, a, false, b0, (short)0, acc0, false, false);
    acc1 = __builtin_amdgcn_wmma_f32_16x16x4_f32(false, a, false, b1, (short)0, acc1, false, false);
    acc2 = __builtin_amdgcn_wmma_f32_16x16x4_f32(false, a, false, b2, (short)0, acc2, false, false);
    acc3 = __builtin_amdgcn_wmma_f32_16x16x4_f32(false, a, false, b3, (short)0, acc3, false, false);
  }

#pragma unroll
  for (int t = 0; t < 4; ++t) {
    v8f acc = (t == 0) ? acc0 : (t == 1) ? acc1 : (t == 2) ? acc2 : acc3;
    const int col = nb + 16 * t + m;
    float bv = bias ? bias[col] : 0.0f;
#pragma unroll
    for (int r = 0; r < 8; ++r) {
      int row = mb + r + 8 * half;
      float v = acc[r] + bv;
      if (act) v = silu_f(v);
      C[(size_t)row * ldc + col] = v;
    }
  }
}

// ---------------------------------------------------------------------------
// s[n][f] = emb[z[n]][f]
// ---------------------------------------------------------------------------
__global__ void embed_kernel(const float* __restrict__ emb,
                             const int* __restrict__ z,
                             float* __restrict__ s)
{
  int n = blockIdx.x, f = threadIdx.x;
  s[(size_t)n * F_DIM + f] = emb[(size_t)z[n] * F_DIM + f];
}

// ---------------------------------------------------------------------------
// Fused message kernel: one block per (molecule b, target atom i), 128 threads
// over feature dim. Computes mask/d/cut/rbf per neighbor j in LDS, stages the
// needed W_rbf columns in LDS, then accumulates
//   delta_s[f] = sum_j phi1[b,j,f] * (rbf_j @ W_rbf[:,f]      + b_rbf[f]     ) * mask
//   t2[x][f]   = sum_j phi3[b,j,f] * (rbf_j @ W_rbf[:,256+f]  + b_rbf[256+f] ) * mask * reld[j][x]
// (t1 is identically zero because vb0 == 0; the F..2F split only feeds t1.)
// Writes snew = s + delta_s, v = t2, and the float mask for later aggregation.
// ---------------------------------------------------------------------------
__global__ void message_kernel(const float* __restrict__ pos,
                               const float* __restrict__ phi,
                               const float* __restrict__ s,
                               const float* __restrict__ W_rbf,
                               const float* __restrict__ b_rbf,
                               float* __restrict__ snew,
                               float* __restrict__ vout,
                               float* __restrict__ maskg)
{
  __shared__ float sh_reld[NPM][3];
  __shared__ float sh_mask[NPM];
  __shared__ float sh_rbf[NPM][N_RBF];
  __shared__ float sh_w1[N_RBF][F_DIM];
  __shared__ float sh_w3[N_RBF][F_DIM];

  const int bi = blockIdx.x;
  const int b  = bi >> 5;
  const int i  = bi & 31;
  const int f  = threadIdx.x;

  // stage W_rbf columns [0:128) and [256:384)
#pragma unroll
  for (int n = 0; n < N_RBF; ++n) {
    sh_w1[n][f] = W_rbf[(size_t)n * F3 + f];
    sh_w3[n][f] = W_rbf[(size_t)n * F3 + 2 * F_DIM + f];
  }

  if (threadIdx.x < NPM) {
    int j = threadIdx.x;
    float px = pos[(size_t)(b * NPM + i) * 3 + 0];
    float py = pos[(size_t)(b * NPM + i) * 3 + 1];
    float pz = pos[(size_t)(b * NPM + i) * 3 + 2];
    float rx = px - pos[(size_t)(b * NPM + j) * 3 + 0];
    float ry = py - pos[(size_t)(b * NPM + j) * 3 + 1];
    float rz = pz - pos[(size_t)(b * NPM + j) * 3 + 2];
    float d2 = rx * rx + ry * ry + rz * rz;
    bool  mk = (d2 <= CUTOFF2) && (j != i);
    float d  = sqrtf(mk ? d2 : 1.0f);
    // cut exactly as in the reference (including the d>0 overwrite)
    float cut = (d <= CUTOFF) ? 0.5f * (__cosf(PI_F * d / CUTOFF) + 1.0f) : d;
    cut = (d > 0.0f) ? 0.0f : cut;
    float inv_d = 1.0f / d;
#pragma unroll
    for (int n = 0; n < N_RBF; ++n)
      sh_rbf[j][n] = cut * __sinf((float)(n + 1) * PI_F * d / CUTOFF) * inv_d;
    sh_reld[j][0] = rx * d;
    sh_reld[j][1] = ry * d;
    sh_reld[j][2] = rz * d;
    float mf = mk ? 1.0f : 0.0f;
    sh_mask[j] = mf;
    maskg[(size_t)bi * NPM + j] = mf;
  }
  __syncthreads();

  const float base1 = b_rbf[f];
  const float base3 = b_rbf[2 * F_DIM + f];
  float ds = 0.0f, tx = 0.0f, ty = 0.0f, tz = 0.0f;

  for (int j = 0; j < NPM; ++j) {
    float mf = sh_mask[j];
    float w1 = base1, w3 = base3;
#pragma unroll
    for (int n = 0; n < N_RBF; ++n) {
      float r = sh_rbf[j][n];
      w1 = fmaf(r, sh_w1[n][f], w1);
      w3 = fmaf(r, sh_w3[n][f], w3);
    }
    const float* ph = phi + (size_t)(b * NPM + j) * F3;
    float p1 = ph[f];
    float p3 = ph[2 * F_DIM + f];
    ds = fmaf(p1 * w1, mf, ds);
    float c3 = p3 * w3 * mf;
    tx = fmaf(c3, sh_reld[j][0], tx);
    ty = fmaf(c3, sh_reld[j][1], ty);
    tz = fmaf(c3, sh_reld[j][2], tz);
  }

  const size_t n = (size_t)(b * NPM + i);
  snew[n * F_DIM + f] = s[n * F_DIM + f] + ds;
  vout[0 * (size_t)NATOMS * F_DIM + n * F_DIM + f] = tx;
  vout[1 * (size_t)NATOMS * F_DIM + n * F_DIM + f] = ty;
  vout[2 * (size_t)NATOMS * F_DIM + n * F_DIM + f] = tz;
}

// ---------------------------------------------------------------------------
// Vn = sqrt(sum_x V^2 + 1e-12);  cat = [snew, Vn]  (N x 256)
// ---------------------------------------------------------------------------
__global__ void vn_cat_kernel(const float* __restrict__ snew,
                              const float* __restrict__ V,
                              float* __restrict__ cat)
{
  size_t n = blockIdx.x;
  int f = threadIdx.x;
  float v0 = V[0 * (size_t)NATOMS * F_DIM + n * F_DIM + f];
  float v1 = V[1 * (size_t)NATOMS * F_DIM + n * F_DIM + f];
  float v2 = V[2 * (size_t)NATOMS * F_DIM + n * F_DIM + f];
  float vn = sqrtf(v0 * v0 + v1 * v1 + v2 * v2 + 1e-12f);
  cat[n * (2 * F_DIM) + f]         = snew[n * F_DIM + f];
  cat[n * (2 * F_DIM) + F_DIM + f] = vn;
}

// ---------------------------------------------------------------------------
// m_s[n][f] = sp[n][F+f] + (sum_x U*V) * sp[n][2F+f]
// ---------------------------------------------------------------------------
__global__ void ms_kernel(const float* __restrict__ sp,
                          const float* __restrict__ U,
                          const float* __restrict__ V,
                          float* __restrict__ m_s)
{
  size_t n = blockIdx.x;
  int f = threadIdx.x;
  size_t o = n * F_DIM + f;
  float dot = U[o] * V[o]
            + U[(size_t)NATOMS * F_DIM + o] * V[(size_t)NATOMS * F_DIM + o]
            + U[2 * (size_t)NATOMS * F_DIM + o] * V[2 * (size_t)NATOMS * F_DIM + o];
  m_s[o] = sp[n * F3 + F_DIM + f] + dot * sp[n * F3 + 2 * F_DIM + f];
}

// ---------------------------------------------------------------------------
// sfin[b,i,f] = snew[b,i,f] + sum_j mask[b,i,j] * m_s[b,j,f]
// ---------------------------------------------------------------------------
__global__ void agg_kernel(const float* __restrict__ maskg,
                           const float* __restrict__ m_s,
                           const float* __restrict__ snew,
                           float* __restrict__ sfin)
{
  const int bi = blockIdx.x;
  const int b  = bi >> 5;
  const int f  = threadIdx.x;
  float acc = 0.0f;
  for (int j = 0; j < NPM; ++j)
    acc = fmaf(maskg[(size_t)bi * NPM + j],
               m_s[(size_t)(b * NPM + j) * F_DIM + f], acc);
  sfin[(size_t)bi * F_DIM + f] = snew[(size_t)bi * F_DIM + f] + acc;
}

// ---------------------------------------------------------------------------
// out[b] = sum_i ( h[b*32+i] . W_o2 + b_o2 )
// ---------------------------------------------------------------------------
__global__ void out_reduce_kernel(const float* __restrict__ h,
                                  const float* __restrict__ W_o2,
                                  const float* __restrict__ b_o2,
                                  float* __restrict__ out)
{
  __shared__ float red[F_DIM];
  const int b = blockIdx.x;
  const int f = threadIdx.x;
  float acc = 0.0f;
  for (int i = 0; i < NPM; ++i)
    acc += h[(size_t)(b * NPM + i) * F_DIM + f];
  red[f] = acc * W_o2[f];
  __syncthreads();
  for (int off = F_DIM / 2; off > 0; off >>= 1) {
    if (f < off) red[f] += red[f + off];
    __syncthreads();
  }
  if (f == 0) out[b] = red[0] + (float)NPM * b_o2[0];
}

// ---------------------------------------------------------------------------
extern "C" void kernel_launch(void* const* d_in, const int* in_sizes, int n_in,
                              void* d_out, int out_size, void* d_ws, size_t ws_size,
                              hipStream_t stream)
{
  // inputs: z, pos, batch, then params in dict order
  const int*   z      = (const int*)  d_in[0];
  const float* pos    = (const float*)d_in[1];
  // d_in[2] = batch (unused; derived as n/32)
  const float* emb    = (const float*)d_in[3];
  const float* W_phi1 = (const float*)d_in[4];
  const float* b_phi1 = (const float*)d_in[5];
  const float* W_phi2 = (const float*)d_in[6];
  const float* b_phi2 = (const float*)d_in[7];
  const float* W_rbf  = (const float*)d_in[8];
  const float* b_rbf  = (const float*)d_in[9];
  const float* W_U    = (const float*)d_in[10];
  const float* W_V    = (const float*)d_in[11];
  const float* W_up1  = (const float*)d_in[12];
  const float* b_up1  = (const float*)d_in[13];
  const float* W_up2  = (const float*)d_in[14];
  const float* b_up2  = (const float*)d_in[15];
  const float* W_o1   = (const float*)d_in[16];
  const float* b_o1   = (const float*)d_in[17];
  const float* W_o2   = (const float*)d_in[18];
  const float* b_o2   = (const float*)d_in[19];
  float* out = (float*)d_out;

  // workspace layout (floats)
  float* ws    = (float*)d_ws;
  const size_t NF  = (size_t)NATOMS * F_DIM;   // 524288
  const size_t NF3 = (size_t)NATOMS * F3;      // 1572864
  float* s_    = ws;             // N x 128
  float* x1    = s_   + NF;      // N x 128   (later reused as st)
  float* phi   = x1   + NF;      // N x 384   (later reused as sp)
  float* snew  = phi  + NF3;     // N x 128
  float* v     = snew + NF;      // 3 x N x 128  (later reused as cat: N x 256)
  float* U     = v    + 3 * NF;  // 3 x N x 128
  float* V     = U    + 3 * NF;  // 3 x N x 128
  float* maskg = V    + 3 * NF;  // N x 32
  float* msb   = maskg + (size_t)NATOMS * NPM; // N x 128 (m_s, later h)
  float* sfin  = msb  + NF;      // N x 128
  (void)ws_size; (void)in_sizes; (void)n_in; (void)out_size;

  const size_t ldsK128 = (size_t)F_DIM * 64 * sizeof(float);      // 32 KB
  const size_t ldsK256 = (size_t)(2 * F_DIM) * 64 * sizeof(float);// 64 KB

  // 1) embedding gather
  embed_kernel<<<NATOMS, F_DIM, 0, stream>>>(emb, z, s_);

  // 2) x1 = silu(s @ W_phi1 + b_phi1)           [4096x128 @ 128x128]
  gemm_wmma_f32<<<dim3(F_DIM/64, NATOMS/64), 128, ldsK128, stream>>>(
      s_, F_DIM, W_phi1, F_DIM, b_phi1, x1, F_DIM, F_DIM, 1);

  // 3) phi = x1 @ W_phi2 + b_phi2               [4096x128 @ 128x384]
  gemm_wmma_f32<<<dim3(F3/64, NATOMS/64), 128, ldsK128, stream>>>(
      x1, F_DIM, W_phi2, F3, b_phi2, phi, F3, F_DIM, 0);

  // 4) fused message stage: snew, v(=t2), mask
  message_kernel<<<NATOMS, F_DIM, 0, stream>>>(
      pos, phi, s_, W_rbf, b_rbf, snew, v, maskg);

  // 5) U = v @ W_U ; 6) V = v @ W_V             [12288x128 @ 128x128]
  gemm_wmma_f32<<<dim3(F_DIM/64, 3*NATOMS/64), 128, ldsK128, stream>>>(
      v, F_DIM, W_U, F_DIM, nullptr, U, F_DIM, F_DIM, 0);
  gemm_wmma_f32<<<dim3(F_DIM/64, 3*NATOMS/64), 128, ldsK128, stream>>>(
      v, F_DIM, W_V, F_DIM, nullptr, V, F_DIM, F_DIM, 0);

  // 7) cat = [snew, ||V||]  (reuse v buffer)
  float* cat = v;
  vn_cat_kernel<<<NATOMS, F_DIM, 0, stream>>>(snew, V, cat);

  // 8) st = silu(cat @ W_up1 + b_up1)           [4096x256 @ 256x128]
  float* st = x1;
  gemm_wmma_f32<<<dim3(F_DIM/64, NATOMS/64), 128, ldsK256, stream>>>(
      cat, 2*F_DIM, W_up1, F_DIM, b_up1, st, F_DIM, 2*F_DIM, 1);

  // 9) sp = st @ W_up2 + b_up2                  [4096x128 @ 128x384]
  float* sp = phi;
  gemm_wmma_f32<<<dim3(F3/64, NATOMS/64), 128, ldsK128, stream>>>(
      st, F_DIM, W_up2, F3, b_up2, sp, F3, F_DIM, 0);

  // 10) m_s = sp[:,F:2F] + (U.V) * sp[:,2F:3F]
  ms_kernel<<<NATOMS, F_DIM, 0, stream>>>(sp, U, V, msb);

  // 11) sfin = snew + mask @ m_s
  agg_kernel<<<NATOMS, F_DIM, 0, stream>>>(maskg, msb, snew, sfin);

  // 12) h = silu(sfin @ W_o1 + b_o1)            [4096x128 @ 128x128]
  float* h = msb;
  gemm_wmma_f32<<<dim3(F_DIM/64, NATOMS/64), 128, ldsK128, stream>>>(
      sfin, F_DIM, W_o1, F_DIM, b_o1, h, F_DIM, F_DIM, 1);

  // 13) per-molecule reduction to the 128 outputs
  out_reduce_kernel<<<NMOL, F_DIM, 0, stream>>>(h, W_o2, b_o2, out);
}